// MagNet_82824149336813
// MI455X (gfx1250) — compile-verified
//
#include <hip/hip_runtime.h>
#include <hip/hip_bf16.h>

#define N_NODES 100000
#define E_EDGES 1600000
#define EMB 32
#define HID 16
#define NUM_SYMP 15
#define BY 4
#define NCOLS (BY + 1 + NUM_SYMP)   // 20 int columns in x
#define TWO_E (2 * E_EDGES)

typedef __attribute__((ext_vector_type(2))) float v2f;
typedef __attribute__((ext_vector_type(8))) float v8f;

// ---------------------------------------------------------------- utilities
__global__ void zero_kernel(float* __restrict__ p, int n) {
    int i = blockIdx.x * blockDim.x + threadIdx.x;
    if (i < n) p[i] = 0.0f;
}

// ---------------------------------------------------------------- embeddings
// h[n][c] = (birth_e + gender_e + mean(symp_e)) / 3
__global__ void embed_kernel(const int* __restrict__ x,
                             const float* __restrict__ birth_tab,
                             const float* __restrict__ gender_tab,
                             const float* __restrict__ symp_tab,
                             float* __restrict__ h) {
    int tid = blockIdx.x * blockDim.x + threadIdx.x;
    if (tid >= N_NODES * EMB) return;
    int n = tid >> 5;      // / EMB
    int c = tid & 31;      // % EMB
    const int* xr = x + (size_t)n * NCOLS;
    int bidx = 0;          // argmax(x[:, :BY]==1): first index equal to 1, default 0
    #pragma unroll
    for (int i = BY - 1; i >= 0; --i) if (xr[i] == 1) bidx = i;
    float be = birth_tab[bidx * EMB + c];
    float ge = gender_tab[xr[BY] * EMB + c];
    float fs = 0.0f;
    #pragma unroll
    for (int i = 0; i < NUM_SYMP; ++i) {
        int v = xr[BY + 1 + i];
        fs += symp_tab[(i * 3 + v) * EMB + c];
    }
    fs *= (1.0f / (float)NUM_SYMP);
    h[(size_t)n * EMB + c] = (be + ge + fs) * (1.0f / 3.0f);
}

// ---------------------------------------------------------------- degree / dinv
__global__ void degree_kernel(const int* __restrict__ src, const int* __restrict__ dst,
                              const float* __restrict__ w, float* __restrict__ deg) {
    int e = blockIdx.x * blockDim.x + threadIdx.x;
    if (e >= E_EDGES) return;
    float hw = 0.5f * w[e];
    unsafeAtomicAdd(&deg[dst[e]], hw);
    unsafeAtomicAdd(&deg[src[e]], hw);
}

__global__ void dinv_kernel(float* __restrict__ deg) {
    int n = blockIdx.x * blockDim.x + threadIdx.x;
    if (n >= N_NODES) return;
    float d = deg[n];
    deg[n] = (d > 0.0f) ? rsqrtf(d) : 0.0f;
}

// ---------------------------------------------------------------- per-edge complex coeffs
// directed edge e in [0, 2E): first half (src->dst, +w), second half (dst->src, -w)
__global__ void coeff_kernel(const int* __restrict__ src, const int* __restrict__ dst,
                             const float* __restrict__ w, const float* __restrict__ dinv,
                             float* __restrict__ a_re, float* __restrict__ a_im) {
    int e = blockIdx.x * blockDim.x + threadIdx.x;
    if (e >= TWO_E) return;
    int s, d; float ww, sgn;
    if (e < E_EDGES) { s = src[e]; d = dst[e]; ww = w[e]; sgn = 1.0f; }
    else             { s = dst[e - E_EDGES]; d = src[e - E_EDGES]; ww = w[e - E_EDGES]; sgn = -1.0f; }
    float norm  = dinv[s] * (0.5f * ww) * dinv[d];
    float theta = sgn * 1.5707963267948966f * ww;  // 2*pi*Q*(+-w), Q=0.25
    a_re[e] = -norm * cosf(theta);
    a_im[e] = -norm * sinf(theta);
}

// ---------------------------------------------------------------- complex SpMM scatter
// y += A x (complex), per (edge, 4-channel group) thread; F is a multiple of 4
template <int F>
__global__ void spmm_kernel(const int* __restrict__ src, const int* __restrict__ dst,
                            const float* __restrict__ a_re, const float* __restrict__ a_im,
                            const float* __restrict__ xr, const float* __restrict__ xi,
                            float* __restrict__ yr, float* __restrict__ yi) {
    const int GRP = F >> 2;
    long long tid = (long long)blockIdx.x * blockDim.x + threadIdx.x;
    if (tid >= (long long)TWO_E * GRP) return;
    int e = (int)(tid / GRP);
    int c = ((int)(tid % GRP)) << 2;
    int s, d;
    if (e < E_EDGES) { s = src[e]; d = dst[e]; }
    else             { s = dst[e - E_EDGES]; d = src[e - E_EDGES]; }
    float ar = a_re[e], ai = a_im[e];
    const float4 r4 = *(const float4*)(xr + (size_t)s * F + c);
    const float4 i4 = *(const float4*)(xi + (size_t)s * F + c);
    float* pr = yr + (size_t)d * F + c;
    float* pi = yi + (size_t)d * F + c;
    unsafeAtomicAdd(pr + 0, ar * r4.x - ai * i4.x);
    unsafeAtomicAdd(pr + 1, ar * r4.y - ai * i4.y);
    unsafeAtomicAdd(pr + 2, ar * r4.z - ai * i4.z);
    unsafeAtomicAdd(pr + 3, ar * r4.w - ai * i4.w);
    unsafeAtomicAdd(pi + 0, ar * i4.x + ai * r4.x);
    unsafeAtomicAdd(pi + 1, ar * i4.y + ai * r4.y);
    unsafeAtomicAdd(pi + 2, ar * i4.z + ai * r4.z);
    unsafeAtomicAdd(pi + 3, ar * i4.w + ai * r4.w);
}

// ---------------------------------------------------------------- stacked-weight prep
// wstk = [W0 - W2 ; W1 ; 2*W2], each K x HID
__global__ void wprep_kernel(const float* __restrict__ W, float* __restrict__ wstk, int K) {
    int i = blockIdx.x * blockDim.x + threadIdx.x;
    int kh = K * HID;
    if (i >= 3 * kh) return;
    int part = i / kh, r = i % kh;
    float w0 = W[r], w1 = W[kh + r], w2 = W[2 * kh + r];
    wstk[i] = (part == 0) ? (w0 - w2) : ((part == 1) ? w1 : 2.0f * w2);
}

// ---------------------------------------------------------------- WMMA GEMM
// out[N][16] = X0@Wa + X1@Wb + X2@Wc + bias, with Wstk = [Wa;Wb;Wc] (each K x 16).
// One wave computes a 16x16 output tile via V_WMMA_F32_16X16X4_F32, K-loop step 4.
template <int K>
__global__ void gemm3_wmma_kernel(const float* __restrict__ X0,
                                  const float* __restrict__ X1,
                                  const float* __restrict__ X2,
                                  const float* __restrict__ Wstk,
                                  const float* __restrict__ bias,
                                  float* __restrict__ out) {
    int wave = (blockIdx.x * blockDim.x + threadIdx.x) >> 5;
    int lane = threadIdx.x & 31;
    const int MT = N_NODES / 16;          // 6250 tiles, exact
    if (wave >= MT) return;               // wave-uniform: EXEC all-ones around WMMA
    int m0   = wave * 16;
    int half = lane >> 4;                 // 0: lanes 0-15, 1: lanes 16-31
    int l    = lane & 15;
    v8f acc = {};
    #pragma unroll
    for (int part = 0; part < 3; ++part) {
        const float* X = (part == 0) ? X0 : (part == 1) ? X1 : X2;
        const float* W = Wstk + part * K * HID;
        #pragma unroll
        for (int k0 = 0; k0 < K; k0 += 4) {
            // A 16x4: row = m0+l; half selects K pair {k0,k0+1} / {k0+2,k0+3}
            const float* ap = X + (size_t)(m0 + l) * K + k0 + half * 2;
            v2f a; a.x = ap[0]; a.y = ap[1];
            // B 4x16: col = l; v0 holds K=k0 (half0) / k0+2 (half1), v1 holds k0+1 / k0+3
            v2f b;
            b.x = W[(k0 + half * 2 + 0) * HID + l];
            b.y = W[(k0 + half * 2 + 1) * HID + l];
            acc = __builtin_amdgcn_wmma_f32_16x16x4_f32(
                false, a, false, b, (short)0, acc, false, false);
        }
    }
    float bb = bias[l];
    #pragma unroll
    for (int v = 0; v < 8; ++v)           // D: vgpr v -> row m0 + v + half*8, col l
        out[(size_t)(m0 + v + half * 8) * HID + l] = acc[v] + bb;
}

// ---------------------------------------------------------------- fused linear head prep
// u[k] = sum_j Wc[j][k] * Wl[j];  c0 = sum_j bc[j]*Wl[j] + bl
__global__ void finprep_kernel(const float* __restrict__ Wc, const float* __restrict__ bc,
                               const float* __restrict__ Wl, const float* __restrict__ bl,
                               float* __restrict__ u, float* __restrict__ c0) {
    int k = threadIdx.x;
    if (k < 2 * HID) {
        float s = 0.0f;
        for (int j = 0; j < 2 * HID; ++j) s += Wc[j * 2 * HID + k] * Wl[j];
        u[k] = s;
    }
    if (k == 0) {
        float s = 0.0f;
        for (int j = 0; j < 2 * HID; ++j) s += bc[j] * Wl[j];
        *c0 = s + bl[0];
    }
}

__global__ void final_kernel(const float* __restrict__ r2, const float* __restrict__ i2,
                             const float* __restrict__ u, const float* __restrict__ c0,
                             float* __restrict__ y) {
    int n = blockIdx.x * blockDim.x + threadIdx.x;
    if (n >= N_NODES) return;
    float s = *c0;
    #pragma unroll
    for (int k = 0; k < HID; ++k) s += r2[(size_t)n * HID + k] * u[k];
    #pragma unroll
    for (int k = 0; k < HID; ++k) s += i2[(size_t)n * HID + k] * u[HID + k];
    y[n] = s;
}

// ---------------------------------------------------------------- launch
static inline int blks(long long n, int b) { return (int)((n + b - 1) / b); }

extern "C" void kernel_launch(void* const* d_in, const int* in_sizes, int n_in,
                              void* d_out, int out_size, void* d_ws, size_t ws_size,
                              hipStream_t stream) {
    const int*   x          = (const int*)  d_in[0];
    const int*   edge_index = (const int*)  d_in[1];
    const float* edge_w     = (const float*)d_in[2];
    const float* birth_tab  = (const float*)d_in[3];
    const float* gender_tab = (const float*)d_in[4];
    const float* symp_tab   = (const float*)d_in[5];
    const float* W1         = (const float*)d_in[6];
    const float* b1         = (const float*)d_in[7];
    const float* W2         = (const float*)d_in[8];
    const float* b2         = (const float*)d_in[9];
    const float* Wc         = (const float*)d_in[10];
    const float* bc         = (const float*)d_in[11];
    const float* Wl         = (const float*)d_in[12];
    const float* bl         = (const float*)d_in[13];
    float* y = (float*)d_out;

    const int* src = edge_index;            // edge_index[0]
    const int* dst = edge_index + E_EDGES;  // edge_index[1]

    // workspace carve-up (floats)
    float* ws = (float*)d_ws;
    size_t off = 0;
    float* h    = ws + off; off += (size_t)N_NODES * EMB;
    float* dinv = ws + off; off += N_NODES;            // deg -> dinv in place
    float* are  = ws + off; off += TWO_E;
    float* aim  = ws + off; off += TWO_E;
    float* t1r  = ws + off; off += (size_t)N_NODES * EMB;
    float* t1i  = ws + off; off += (size_t)N_NODES * EMB;
    float* t2r  = ws + off; off += (size_t)N_NODES * EMB;
    float* t2i  = ws + off; off += (size_t)N_NODES * EMB;
    float* r1   = ws + off; off += (size_t)N_NODES * HID;
    float* i1   = ws + off; off += (size_t)N_NODES * HID;
    float* r2   = ws + off; off += (size_t)N_NODES * HID;
    float* i2   = ws + off; off += (size_t)N_NODES * HID;
    float* wstk = ws + off; off += 3 * EMB * HID;
    float* u    = ws + off; off += 2 * HID;
    float* c0   = ws + off; off += 1;

    const int B = 256;

    // features + normalization
    zero_kernel<<<blks(N_NODES, B), B, 0, stream>>>(dinv, N_NODES);
    embed_kernel<<<blks((long long)N_NODES * EMB, B), B, 0, stream>>>(
        x, birth_tab, gender_tab, symp_tab, h);
    degree_kernel<<<blks(E_EDGES, B), B, 0, stream>>>(src, dst, edge_w, dinv);
    dinv_kernel<<<blks(N_NODES, B), B, 0, stream>>>(dinv);
    coeff_kernel<<<blks(TWO_E, B), B, 0, stream>>>(src, dst, edge_w, dinv, are, aim);

    // ---- conv layer 1 (K = EMB = 32), input h (re == im == h)
    zero_kernel<<<blks((long long)N_NODES * EMB, B), B, 0, stream>>>(t1r, N_NODES * EMB);
    zero_kernel<<<blks((long long)N_NODES * EMB, B), B, 0, stream>>>(t1i, N_NODES * EMB);
    spmm_kernel<EMB><<<blks((long long)TWO_E * (EMB / 4), B), B, 0, stream>>>(
        src, dst, are, aim, h, h, t1r, t1i);
    zero_kernel<<<blks((long long)N_NODES * EMB, B), B, 0, stream>>>(t2r, N_NODES * EMB);
    zero_kernel<<<blks((long long)N_NODES * EMB, B), B, 0, stream>>>(t2i, N_NODES * EMB);
    spmm_kernel<EMB><<<blks((long long)TWO_E * (EMB / 4), B), B, 0, stream>>>(
        src, dst, are, aim, t1r, t1i, t2r, t2i);
    wprep_kernel<<<blks(3 * EMB * HID, B), B, 0, stream>>>(W1, wstk, EMB);
    {
        int waves = N_NODES / 16;
        gemm3_wmma_kernel<EMB><<<blks((long long)waves * 32, B), B, 0, stream>>>(
            h, t1r, t2r, wstk, b1, r1);
        gemm3_wmma_kernel<EMB><<<blks((long long)waves * 32, B), B, 0, stream>>>(
            h, t1i, t2i, wstk, b1, i1);
    }

    // ---- conv layer 2 (K = HID = 16), input (r1, i1)
    zero_kernel<<<blks((long long)N_NODES * HID, B), B, 0, stream>>>(t1r, N_NODES * HID);
    zero_kernel<<<blks((long long)N_NODES * HID, B), B, 0, stream>>>(t1i, N_NODES * HID);
    spmm_kernel<HID><<<blks((long long)TWO_E * (HID / 4), B), B, 0, stream>>>(
        src, dst, are, aim, r1, i1, t1r, t1i);
    zero_kernel<<<blks((long long)N_NODES * HID, B), B, 0, stream>>>(t2r, N_NODES * HID);
    zero_kernel<<<blks((long long)N_NODES * HID, B), B, 0, stream>>>(t2i, N_NODES * HID);
    spmm_kernel<HID><<<blks((long long)TWO_E * (HID / 4), B), B, 0, stream>>>(
        src, dst, are, aim, t1r, t1i, t2r, t2i);
    wprep_kernel<<<blks(3 * HID * HID, B), B, 0, stream>>>(W2, wstk, HID);
    {
        int waves = N_NODES / 16;
        gemm3_wmma_kernel<HID><<<blks((long long)waves * 32, B), B, 0, stream>>>(
            r1, t1r, t2r, wstk, b2, r2);
        gemm3_wmma_kernel<HID><<<blks((long long)waves * 32, B), B, 0, stream>>>(
            i1, t1i, t2i, wstk, b2, i2);
    }

    // ---- fused linear head: y = [r2|i2] @ u + c0
    finprep_kernel<<<1, 32, 0, stream>>>(Wc, bc, Wl, bl, u, c0);
    final_kernel<<<blks(N_NODES, B), B, 0, stream>>>(r2, i2, u, c0, y);
}